// single_head_61649960567444
// MI455X (gfx1250) — compile-verified
//
#include <hip/hip_runtime.h>
#include <math.h>

#define C_DIM 128
#define NEG_SLOPE 0.01f
#define LDS_STRIDE 36   // padded row stride (floats): 16B-aligned rows, conflict-free banks

typedef __attribute__((ext_vector_type(2))) float v2f;
typedef __attribute__((ext_vector_type(8))) float v8f;

// ---------------- 1) per-node score dot: s[i] = x[i] . w  (wave per node) ---
__global__ void __launch_bounds__(256) k_dot(const float* __restrict__ x,
                                             const float* __restrict__ w,
                                             float* __restrict__ s,
                                             float* __restrict__ pmax, int N) {
  const int wid = threadIdx.x >> 5, lane = threadIdx.x & 31;
  const int node = blockIdx.x * 8 + wid;
  __shared__ float smax[8];
  float sum = 0.f;
  if (node < N) {
    const float4 xv = *(const float4*)(x + (size_t)node * C_DIM + lane * 4);
    const float4 wv = *(const float4*)(w + lane * 4);
    sum = xv.x * wv.x + xv.y * wv.y + xv.z * wv.z + xv.w * wv.w;
    for (int off = 16; off >= 1; off >>= 1) sum += __shfl_down(sum, off, 32);
  }
  if (lane == 0) {
    if (node < N) s[node] = sum;
    smax[wid] = (node < N) ? sum : -3.4e38f;
  }
  __syncthreads();
  if (threadIdx.x == 0) {
    float m = smax[0];
    for (int i = 1; i < 8; ++i) m = fmaxf(m, smax[i]);
    pmax[blockIdx.x] = m;
  }
}

// ---------------- 2) reduce block maxima -> scal[0] = m ---------------------
__global__ void __launch_bounds__(256) k_max(const float* __restrict__ pmax,
                                             int n, float* __restrict__ scal) {
  __shared__ float sh[256];
  float m = -3.4e38f;
  for (int i = threadIdx.x; i < n; i += 256) m = fmaxf(m, pmax[i]);
  sh[threadIdx.x] = m;
  __syncthreads();
  for (int off = 128; off >= 1; off >>= 1) {
    if ((int)threadIdx.x < off) sh[threadIdx.x] = fmaxf(sh[threadIdx.x], sh[threadIdx.x + off]);
    __syncthreads();
  }
  if (threadIdx.x == 0) scal[0] = sh[0];
}

// ---------------- 3) e[i] = exp(s[i]-m), per-block sums ---------------------
__global__ void __launch_bounds__(256) k_exp(float* __restrict__ s,
                                             const float* __restrict__ scal,
                                             float* __restrict__ psum, int N) {
  __shared__ float sh[256];
  const int i = blockIdx.x * 256 + threadIdx.x;
  const float m = scal[0];
  float e = 0.f;
  if (i < N) { e = expf(s[i] - m); s[i] = e; }
  sh[threadIdx.x] = e;
  __syncthreads();
  for (int off = 128; off >= 1; off >>= 1) {
    if ((int)threadIdx.x < off) sh[threadIdx.x] += sh[threadIdx.x + off];
    __syncthreads();
  }
  if (threadIdx.x == 0) psum[blockIdx.x] = sh[0];
}

// ---------------- 4) Z, invZ, threshold -------------------------------------
__global__ void k_final(const float* __restrict__ psum, int n, float* __restrict__ scal) {
  float Z = 0.f;
  for (int i = 0; i < n; ++i) Z += psum[i];
  const float invZ = 1.f / Z;
  scal[1] = Z;
  scal[2] = invZ;
  // max(softmax) = exp(m-m)/Z = invZ ; thr = min(max - TOL, MIN_SCORE)
  scal[3] = fminf(invZ - 1e-7f, 0.7f);
}

// ---------------- 5) gate/keep per node -------------------------------------
__global__ void __launch_bounds__(256) k_gate(const float* __restrict__ e,
                                              const float* __restrict__ scal,
                                              float* __restrict__ gate,
                                              float* __restrict__ keepf, int N) {
  const int i = blockIdx.x * 256 + threadIdx.x;
  if (i >= N) return;
  const float sc = e[i] * scal[2];
  const bool k = sc > scal[3];
  gate[i] = k ? sc : 0.f;
  keepf[i] = k ? 1.f : 0.f;
}

// ---------------- 6) edge scatter: agg[dst] += x[src]*gate[src] -------------
__global__ void __launch_bounds__(256) k_edge(const int* __restrict__ ei,
                                              const float* __restrict__ x,
                                              const float* __restrict__ gate,
                                              const float* __restrict__ keepf,
                                              float* __restrict__ agg, int E) {
  const int e = blockIdx.x * 8 + (threadIdx.x >> 5);
  if (e >= E) return;
  const int lane = threadIdx.x & 31;
  const int src = ei[e];
  const int dst = ei[E + e];
  const float gs = gate[src];
  if (gs == 0.f) return;          // src dropped -> message is zero
  if (keepf[dst] == 0.f) return;  // dst dropped -> output row masked anyway
  const float4 v = *(const float4*)(x + (size_t)src * C_DIM + lane * 4);
  float* ap = agg + (size_t)dst * C_DIM + lane * 4;
  unsafeAtomicAdd(ap + 0, v.x * gs);
  unsafeAtomicAdd(ap + 1, v.y * gs);
  unsafeAtomicAdd(ap + 2, v.z * gs);
  unsafeAtomicAdd(ap + 3, v.w * gs);
}

// ---------------- 7) fused dual WMMA GEMM + epilogue ------------------------
// out_row = leaky( agg@Wrel^T + b + (x*gate)@Wroot^T ) * keep ; concat prev.
// One wave per 16-node tile; 8 v8f accumulators cover the 128 output columns.
// W_rel/W_root staged in LDS in 32-wide K chunks; B frags via ds_load_b64.
__global__ void __launch_bounds__(256) k_gemm(
    const float* __restrict__ agg, const float* __restrict__ x,
    const float* __restrict__ gate, const float* __restrict__ keepf,
    const float* __restrict__ Wrel, const float* __restrict__ Wroot,
    const float* __restrict__ brel, const float* __restrict__ prev,
    float* __restrict__ out, int N, int ntiles) {
  __shared__ float ldsR[C_DIM * LDS_STRIDE];   // W_rel  chunk  [n=128][k=32 pad 36]
  __shared__ float ldsO[C_DIM * LDS_STRIDE];   // W_root chunk
  __shared__ int blockAny;

  const int wid = threadIdx.x >> 5, lane = threadIdx.x & 31;
  const int tile = blockIdx.x * 8 + wid;
  const bool valid = tile < ntiles;
  const int m0 = valid ? tile * 16 : 0;
  const int hi = lane >> 4;   // lane half: K pair select (A/B), M+8 select (C/D)
  const int rn = lane & 15;   // A row / B,C column within tile

  if (threadIdx.x == 0) blockAny = 0;
  __syncthreads();

  // always copy new_x_prev into the first 128 output columns
  if (valid) {
    for (int r = 0; r < 16; ++r) {
      const int row = m0 + r;
      if (row < N) {
        const float4 pv = *(const float4*)(prev + (size_t)row * C_DIM + lane * 4);
        *(float4*)(out + (size_t)row * 2 * C_DIM + lane * 4) = pv;
      }
    }
  }

  const int arow = m0 + rn;
  const int arowc = (arow < N) ? arow : 0;   // clamped: rows>=N never stored
  const float kflag = (valid && arow < N) ? keepf[arow] : 0.f;
  if (kflag != 0.f) blockAny = 1;            // benign race: all writers store 1
  __syncthreads();

  if (!blockAny) {                           // block-uniform: no barrier hazard
    if (valid) {
      const float4 z = {0.f, 0.f, 0.f, 0.f};
      for (int r = 0; r < 16; ++r) {
        const int row = m0 + r;
        if (row < N) *(float4*)(out + (size_t)row * 2 * C_DIM + C_DIM + lane * 4) = z;
      }
    }
    return;
  }

  const float gr = (valid && arow < N) ? gate[arow] : 0.f;
  const float* aggRow = agg + (size_t)arowc * C_DIM + 2 * hi;
  const float* xRow   = x   + (size_t)arowc * C_DIM + 2 * hi;

  v8f c[8];
#pragma unroll
  for (int nt = 0; nt < 8; ++nt) c[nt] = (v8f){0.f, 0.f, 0.f, 0.f, 0.f, 0.f, 0.f, 0.f};

  for (int c0 = 0; c0 < C_DIM; c0 += 32) {   // 4 K-chunks of 32
    __syncthreads();                          // LDS reuse guard
    // cooperative staging: 128 rows x 8 quads per matrix, 256 threads
    for (int u = threadIdx.x; u < C_DIM * 8; u += 256) {
      const int row = u >> 3, q = u & 7;
      const float4 vR = *(const float4*)(Wrel  + (size_t)row * C_DIM + c0 + q * 4);
      const float4 vO = *(const float4*)(Wroot + (size_t)row * C_DIM + c0 + q * 4);
      *(float4*)(&ldsR[row * LDS_STRIDE + q * 4]) = vR;
      *(float4*)(&ldsO[row * LDS_STRIDE + q * 4]) = vO;
    }
    // preload this chunk's A fragments (independent of LDS -> overlaps staging)
    v2f aA[8], aX[8];
#pragma unroll
    for (int kk = 0; kk < 8; ++kk) {
      aA[kk] = *(const v2f*)(aggRow + c0 + kk * 4);
      aX[kk] = *(const v2f*)(xRow + c0 + kk * 4) * gr;   // x_p frag = x * gate
    }
    __syncthreads();

#pragma unroll
    for (int kk = 0; kk < 8; ++kk) {
      v2f bR[8], bO[8];                       // distinct regs -> load clause, no wait-0 chain
#pragma unroll
      for (int nt = 0; nt < 8; ++nt) {
        const int ng = nt * 16 + rn;          // output column; B[k][n] = W[n][k]
        bR[nt] = *(const v2f*)(&ldsR[ng * LDS_STRIDE + kk * 4 + 2 * hi]);
        bO[nt] = *(const v2f*)(&ldsO[ng * LDS_STRIDE + kk * 4 + 2 * hi]);
      }
#pragma unroll
      for (int nt = 0; nt < 8; ++nt) {
        c[nt] = __builtin_amdgcn_wmma_f32_16x16x4_f32(false, aA[kk], false, bR[nt], (short)0, c[nt], false, false);
        c[nt] = __builtin_amdgcn_wmma_f32_16x16x4_f32(false, aX[kk], false, bO[nt], (short)0, c[nt], false, false);
      }
    }
  }

  if (valid) {
#pragma unroll
    for (int nt = 0; nt < 8; ++nt) {
      const float bias = brel[nt * 16 + rn];
#pragma unroll
      for (int v = 0; v < 8; ++v) {
        const int row = m0 + v + 8 * hi;      // C/D layout: M = v + 8*(lane/16)
        if (row >= N) continue;
        float o = c[nt][v] + bias;
        o = (o >= 0.f) ? o : NEG_SLOPE * o;
        o *= keepf[row];
        out[(size_t)row * 2 * C_DIM + C_DIM + nt * 16 + rn] = o;
      }
    }
  }
}

// ---------------------------------------------------------------------------
extern "C" void kernel_launch(void* const* d_in, const int* in_sizes, int n_in,
                              void* d_out, int out_size, void* d_ws, size_t ws_size,
                              hipStream_t stream) {
  (void)n_in; (void)out_size; (void)ws_size;
  const float* x      = (const float*)d_in[0];
  const int*   ei     = (const int*)d_in[1];
  const float* prev   = (const float*)d_in[2];
  const float* wscore = (const float*)d_in[3];
  const float* Wrel   = (const float*)d_in[4];
  const float* brel   = (const float*)d_in[5];
  const float* Wroot  = (const float*)d_in[6];
  float* out = (float*)d_out;

  const int N = in_sizes[0] / C_DIM;
  const int E = in_sizes[1] / 2;

  // workspace layout (floats)
  float* base  = (float*)d_ws;
  float* s     = base;                 // N  (scores, then exp values)
  float* gate  = base + (size_t)N;     // N
  float* keepf = base + (size_t)2 * N; // N
  float* part  = base + (size_t)3 * N; // block partials
  const int nbDot  = (N + 7) / 8;
  const int nbElem = (N + 255) / 256;
  const int npart  = (nbDot > nbElem) ? nbDot : nbElem;
  float* scal = part + (((size_t)npart + 63) & ~(size_t)63); // m, Z, invZ, thr
  size_t aggOff = (size_t)(scal + 64 - base);
  aggOff = (aggOff + 63) & ~(size_t)63;
  float* agg = base + aggOff;          // N*128 accumulator

  hipMemsetAsync(agg, 0, (size_t)N * C_DIM * sizeof(float), stream);

  k_dot<<<nbDot, 256, 0, stream>>>(x, wscore, s, part, N);
  k_max<<<1, 256, 0, stream>>>(part, nbDot, scal);
  k_exp<<<nbElem, 256, 0, stream>>>(s, scal, part, N);
  k_final<<<1, 1, 0, stream>>>(part, nbElem, scal);
  k_gate<<<nbElem, 256, 0, stream>>>(s, scal, gate, keepf, N);
  k_edge<<<(E + 7) / 8, 256, 0, stream>>>(ei, x, gate, keepf, agg, E);

  const int ntiles = (N + 15) / 16;
  k_gemm<<<(ntiles + 7) / 8, 256, 0, stream>>>(agg, x, gate, keepf, Wrel, Wroot,
                                               brel, prev, out, N, ntiles);
}